// _SingleCodebookQuantizer_85564338471364
// MI455X (gfx1250) — compile-verified
//
#include <hip/hip_runtime.h>

typedef float v2f __attribute__((ext_vector_type(2)));
typedef float v8f __attribute__((ext_vector_type(8)));
typedef unsigned int v4u __attribute__((ext_vector_type(4)));
typedef int v4i __attribute__((ext_vector_type(4)));
typedef int v8i __attribute__((ext_vector_type(8)));

#define DIM 64
#define CHUNK_COLS 64
#define LDS_STRIDE 68  // 64 + 4 pad: 16B-aligned columns, conflict-free b64 lane access

#if __has_builtin(__builtin_amdgcn_tensor_load_to_lds) && __has_builtin(__builtin_amdgcn_s_wait_tensorcnt)
#define USE_TDM 1
#else
#define USE_TDM 0
#endif

#if USE_TDM
// One TDM descriptor moves a 64x64 f32 codebook tile Global->LDS, inserting a
// 4-DWORD pad every 64 DWORDs so the LDS image lands in the stride-68 layout
// the WMMA B-fragment reads expect (ISA 8.3/8.4: pad_interval=5 -> 64 DW,
// pad_amount=3 -> 4 DW).
__device__ __forceinline__ void tdm_load_chunk(const float* gsrc, void* ldsDst)
{
    const unsigned long long ga = (unsigned long long)(size_t)gsrc;
    const unsigned lds = (unsigned)(size_t)ldsDst;  // low 32 bits of shared aperture = LDS byte offset

    v4u g0;
    g0[0] = 1u;                                                // count=1, is_restore=0, gather off
    g0[1] = lds;                                               // lds_addr (bytes)
    g0[2] = (unsigned)(ga & 0xffffffffu);                      // global_addr[31:0]
    g0[3] = (unsigned)((ga >> 32) & 0x1ffffffu) | (2u << 30);  // global_addr[56:32] | type=2

    v8i g1;
    g1[0] = (int)((2u << 16) | (1u << 20) | (5u << 22) | (3u << 25)); // 4B elems; pad: every 64 DW add 4 DW
    g1[1] = (int)(64u << 16);         // tensor_dim0 = 64
    g1[2] = 0;                        // tensor_dim1[15:0] (dim1 = 0x10000: no OOB for our tiles)
    g1[3] = (int)(1u | (64u << 16));  // tensor_dim1[31:16]=1, tile_dim0 = 64
    g1[4] = 64;                       // tile_dim1 = 64 rows, tile_dim2 = 0
    g1[5] = 64;                       // tensor_dim0_stride = 64 elements
    g1[6] = 0;                        // stride hi / dim1_stride lo
    g1[7] = 0;

    const v4i z4 = {0, 0, 0, 0};
#if defined(__clang_major__) && __clang_major__ >= 23
    const v8i z8 = {0, 0, 0, 0, 0, 0, 0, 0};
    __builtin_amdgcn_tensor_load_to_lds(g0, g1, z4, z4, z8, 0);
#else
    __builtin_amdgcn_tensor_load_to_lds(g0, g1, z4, z4, 0);
#endif
}
#endif

// Each wave: 16 rows of x vs. all K codebook entries via V_WMMA_F32_16X16X4_F32.
// score = ||c||^2 - 2*x.c  (||x||^2 constant per row -> dropped for argmin).
// Codebook tiles stream Global->LDS through the TDM, double-buffered so the
// DMA for chunk i+1 overlaps the 64 WMMAs of chunk i.
__global__ __launch_bounds__(256) void vq_wmma_kernel(const float* __restrict__ x,
                                                      const float* __restrict__ cb,
                                                      float* __restrict__ out,
                                                      int N, int K)
{
    __shared__ float cbLds[2][CHUNK_COLS * LDS_STRIDE];
    __shared__ float cbsqLds[2][CHUNK_COLS];

    const int tid  = threadIdx.x;
    const int lane = tid & 31;
    const int wave = tid >> 5;
    const int n16  = lane & 15;
    const int half = lane >> 4;        // 0: lanes 0-15, 1: lanes 16-31

    const int rowBase = blockIdx.x * 128 + wave * 16;
    const int row     = rowBase + n16;

    // A-fragments: x[row][0..63] as 16 K-chunks of 4 (ISA 16x4 f32 A layout:
    // VGPR0 = K0 (lanes 0-15) / K2 (lanes 16-31), VGPR1 = K1 / K3)
    v2f a[16];
#pragma unroll
    for (int c = 0; c < 16; ++c) {
        const int kb = 4 * c + 2 * half;
        a[c] = *(const v2f*)(x + (size_t)row * DIM + kb);
    }

    float best[8];
    int   bidx[8];
#pragma unroll
    for (int g = 0; g < 8; ++g) { best[g] = __builtin_huge_valf(); bidx[g] = 0; }

    const int nChunks = K / CHUNK_COLS;

    // Prologue: chunk 0 -> buffer 0.
#if USE_TDM
    if (wave == 0) {
        tdm_load_chunk(cb, &cbLds[0][0]);
        __builtin_amdgcn_s_wait_tensorcnt(0);
    }
#else
    for (int i = tid; i < CHUNK_COLS * (DIM / 4); i += 256) {
        const int col = i >> 4;
        const int d4  = (i & 15) << 2;
        const float4 v = *(const float4*)(cb + (size_t)col * DIM + d4);
        *(float4*)(&cbLds[0][col * LDS_STRIDE + d4]) = v;
    }
#endif
    __syncthreads();

    int p = 0;
    for (int chunk = 0; chunk < nChunks; ++chunk) {
        const int colBase = chunk * CHUNK_COLS;
        const int nb = p ^ 1;

        // Kick off the next chunk's DMA into the other buffer (overlaps compute).
        if (chunk + 1 < nChunks) {
#if USE_TDM
            if (wave == 0)
                tdm_load_chunk(cb + (size_t)(colBase + CHUNK_COLS) * DIM, &cbLds[nb][0]);
#else
            for (int i = tid; i < CHUNK_COLS * (DIM / 4); i += 256) {
                const int col = i >> 4;
                const int d4  = (i & 15) << 2;
                const float4 v = *(const float4*)(cb + (size_t)(colBase + CHUNK_COLS + col) * DIM + d4);
                *(float4*)(&cbLds[nb][col * LDS_STRIDE + d4]) = v;
            }
#endif
        }

        // Column squared norms for the ready buffer (one thread per column).
        if (tid < CHUNK_COLS) {
            float s = 0.f;
#pragma unroll
            for (int d = 0; d < DIM; ++d) {
                const float v = cbLds[p][tid * LDS_STRIDE + d];
                s += v * v;
            }
            cbsqLds[p][tid] = s;
        }
        __syncthreads();   // cbsq ready

        // 4 column-tiles of 16; D = A(16x64) x B(64x16) via 16 chained x4 WMMAs.
        // B-fragments are preloaded into registers so the 8 paired
        // ds_load_2addr_b64 issue as a clause and the WMMA chain runs
        // without per-load s_wait_dscnt stalls.
        for (int t = 0; t < CHUNK_COLS / 16; ++t) {
            const int tBase = t * 16;
            const float* bcol = &cbLds[p][(tBase + n16) * LDS_STRIDE + 2 * half];

            v2f bfrag[16];
#pragma unroll
            for (int c = 0; c < 16; ++c)
                bfrag[c] = *(const v2f*)(bcol + 4 * c);

            const float cbsq = cbsqLds[p][tBase + n16];

            v8f acc = {};
#pragma unroll
            for (int c = 0; c < 16; ++c)
                acc = __builtin_amdgcn_wmma_f32_16x16x4_f32(
                        /*neg_a=*/false, a[c], /*neg_b=*/false, bfrag[c],
                        /*c_mod=*/(short)0, acc, /*reuse_a=*/false, /*reuse_b=*/false);

            const int gcol = colBase + tBase + n16;
#pragma unroll
            for (int g = 0; g < 8; ++g) {
                const float sc = cbsq - 2.0f * acc[g];
                if (sc < best[g]) { best[g] = sc; bidx[g] = gcol; }  // strict < keeps first occurrence
            }
        }

#if USE_TDM
        // Only the issuing wave waits on its TENSORcnt; barrier releases the rest.
        if (wave == 0 && chunk + 1 < nChunks)
            __builtin_amdgcn_s_wait_tensorcnt(0);
#endif
        __syncthreads();   // next buffer ready, this buffer free for reuse
        p = nb;
    }

    // Reduce (score,index) across the 16 lanes holding each row's columns.
#pragma unroll
    for (int g = 0; g < 8; ++g) {
        float s  = best[g];
        int   bi = bidx[g];
#pragma unroll
        for (int m = 1; m < 16; m <<= 1) {
            const float os = __shfl_xor(s, m, 32);
            const int   oi = __shfl_xor(bi, m, 32);
            if (os < s || (os == s && oi < bi)) { s = os; bi = oi; }
        }
        best[g] = s; bidx[g] = bi;
    }

    // Gather winning codebook rows. Slot g: row rowBase+g lives in lanes 0-15,
    // row rowBase+8+g in lanes 16-31. Broadcast each index, all 32 lanes copy
    // 2 floats -> coalesced 8B stores.
#pragma unroll
    for (int g = 0; g < 8; ++g) {
        const int iA = __shfl(bidx[g], 0, 32);
        const int iB = __shfl(bidx[g], 16, 32);
        const v2f vA = *(const v2f*)(cb + (size_t)iA * DIM + 2 * lane);
        *(v2f*)(out + (size_t)(rowBase + g) * DIM + 2 * lane) = vA;
        const v2f vB = *(const v2f*)(cb + (size_t)iB * DIM + 2 * lane);
        *(v2f*)(out + (size_t)(rowBase + 8 + g) * DIM + 2 * lane) = vB;
    }
}

extern "C" void kernel_launch(void* const* d_in, const int* in_sizes, int n_in,
                              void* d_out, int out_size, void* d_ws, size_t ws_size,
                              hipStream_t stream) {
    const float* x  = (const float*)d_in[0];   // [8,4096,64] f32
    const float* cb = (const float*)d_in[1];   // [8192,64]   f32
    float* out = (float*)d_out;                // [8,4096,64] f32

    const int N = in_sizes[0] / DIM;           // 32768 rows
    const int K = in_sizes[1] / DIM;           // 8192 codes

    vq_wmma_kernel<<<N / 128, 256, 0, stream>>>(x, cb, out, N, K);
}